// ScaledDotProductAttention_13692355740194
// MI455X (gfx1250) — compile-verified
//
#include <hip/hip_runtime.h>

// ---------------------------------------------------------------------------
// Fused masked-prior attention for MI455X (gfx1250, wave32).
//   attn[n,l] = softmax_l( mask ? -1e10 : (q[n,l]·k[n,l]) * r_pri[n%B,l]/8 )
//   out[n,d]  = sum_l attn[n,l] * v[n,l,d]
// N=65536 rows, L=20 neighbors, D=64 features.
//
// HBM-bound (~1 GB streamed, ~0.33 FLOP/B => ~45us floor at 23.3 TB/s).
// Strategy: one wave per row; V tile prefetched to LDS via the CDNA5 async
// engine (global_load_async_to_lds_b128, ASYNCcnt) overlapped with the q·k
// score computation; p×V done with V_WMMA_F32_16X16X4_F32 (f32-exact).
// ---------------------------------------------------------------------------

typedef __attribute__((ext_vector_type(2))) float v2f;
typedef __attribute__((ext_vector_type(8))) float v8f;

#define L_NB 20
#define D_HD 64
#define ROW_F (L_NB * D_HD)          // 1280 floats per (n) tile
#define WAVES_PER_BLOCK 8
#define NEG_INF_F (-1.0e10f)

__global__ __launch_bounds__(256)
void sdpa_prior_kernel(const float* __restrict__ q,
                       const float* __restrict__ k,
                       const float* __restrict__ v,
                       const float* __restrict__ r_pri,
                       const unsigned char* __restrict__ mask,
                       float* __restrict__ out,       // [N, 64]
                       float* __restrict__ attn_out,  // [N, 20]
                       int N, int B)
{
    __shared__ float v_sh[WAVES_PER_BLOCK * ROW_F]; // 40 KB: V tiles
    __shared__ float p_sh[WAVES_PER_BLOCK * 32];    // 1 KB: softmax probs

    const int tid  = threadIdx.x;
    const int wave = tid >> 5;
    const int lane = tid & 31;
    const int n    = blockIdx.x * WAVES_PER_BLOCK + wave;
    if (n >= N) return;                         // wave-uniform guard

    const size_t rowbase = (size_t)n * ROW_F;
    float* vtile = &v_sh[wave * ROW_F];
    float* ptile = &p_sh[wave * 32];

    // ---- Phase 0: async prefetch V[n] (5 KB) into LDS, 16 B per lane per op.
    // Tracked by ASYNCcnt; overlaps with the score computation below.
    {
        const float* gsrc = v + rowbase;
        #pragma unroll
        for (int i = 0; i < 10; ++i) {
            const int f4 = lane + 32 * i;                   // float4 index
            const float* gaddr = gsrc + 4 * f4;
            const unsigned ldst = (unsigned)(size_t)(vtile + 4 * f4);
            asm volatile("global_load_async_to_lds_b128 %0, %1, off"
                         :: "v"(ldst), "v"(gaddr) : "memory");
        }
    }

    // ---- Phase 1: scores s[l] = q[n,l,:]·k[n,l,:] via b128 streaming loads.
    // float4 index f = lane + 32*i  =>  l = 2*i + (lane>=16).
    const float4* q4 = (const float4*)(q + rowbase);
    const float4* k4 = (const float4*)(k + rowbase);
    float a[10];
    #pragma unroll
    for (int i = 0; i < 10; ++i) {
        const int f = lane + 32 * i;
        const float4 qq = q4[f];
        const float4 kk = k4[f];
        a[i] = qq.x * kk.x + qq.y * kk.y + qq.z * kk.z + qq.w * kk.w;
    }
    // Reduce each l across its 16-lane half (xor 1,2,4,8 stays inside half).
    #pragma unroll
    for (int i = 0; i < 10; ++i) {
        float s = a[i];
        s += __shfl_xor(s, 1, 32);
        s += __shfl_xor(s, 2, 32);
        s += __shfl_xor(s, 4, 32);
        s += __shfl_xor(s, 8, 32);
        a[i] = s;
    }

    // ---- Phase 2: prior reweight, mask, stable softmax over L=20.
    const int lpar = (lane >> 4) & 1;           // 0: even l, 1: odd l
    const int b    = n % B;                     // folded-head batch index
    const size_t mrow = (size_t)n * L_NB;
    const size_t rrow = (size_t)b * L_NB;
    float mx = -3.0e38f;
    #pragma unroll
    for (int i = 0; i < 10; ++i) {
        const int l = 2 * i + lpar;
        float s = a[i] * r_pri[rrow + l] * 0.125f;   // / TEMPERATURE
        s = mask[mrow + l] ? NEG_INF_F : s;
        a[i] = s;
        mx = fmaxf(mx, s);
    }
    mx = fmaxf(mx, __shfl_xor(mx, 16, 32));     // combine halves
    float sum = 0.f;
    #pragma unroll
    for (int i = 0; i < 10; ++i) {
        const float e = __expf(a[i] - mx);
        a[i] = e;
        sum += e;
    }
    sum += __shfl_xor(sum, 16, 32);
    const float inv = 1.0f / sum;
    #pragma unroll
    for (int i = 0; i < 10; ++i) a[i] *= inv;

    // Stage probabilities in LDS (lane 0 holds all even l, lane 16 all odd l).
    if (lane == 0 || lane == 16) {
        #pragma unroll
        for (int i = 0; i < 10; ++i) ptile[2 * i + lpar] = a[i];
    }
    asm volatile("s_wait_dscnt 0" ::: "memory");

    // attn output (second tuple element), coalesced 20-float row.
    if (lane < L_NB) attn_out[mrow + lane] = ptile[lane];

    float pl[L_NB];
    #pragma unroll
    for (int l = 0; l < L_NB; ++l) pl[l] = ptile[l];

    // ---- Phase 3: out = p × V via V_WMMA_F32_16X16X4_F32.
    // A (16x4 f32): lanes0-15 hold K=0/1, lanes16-31 hold K=2/3; only row M=0
    //   (lane 0 / lane 16) carries p, other rows zero -> D rows 1..15 ignored.
    // B (4x16 f32): VGPR0 = rows K=0 (lanes0-15) / K=2 (lanes16-31),
    //               VGPR1 = rows K=1 / K=3; columns j = lane&15, from LDS V.
    asm volatile("s_wait_asynccnt 0" ::: "memory");   // V tile resident

    v8f acc[4] = {v8f{}, v8f{}, v8f{}, v8f{}};
    const int j = lane & 15;
    #pragma unroll
    for (int l0 = 0; l0 < L_NB; l0 += 4) {
        const float ax = (lane == 0) ? pl[l0]     : ((lane == 16) ? pl[l0 + 2] : 0.f);
        const float ay = (lane == 0) ? pl[l0 + 1] : ((lane == 16) ? pl[l0 + 3] : 0.f);
        const v2f A = {ax, ay};
        const float* vr = vtile + (l0 + 2 * lpar) * D_HD;  // K rows for this half
        #pragma unroll
        for (int t = 0; t < 4; ++t) {                      // 4 tiles of 16 cols
            const v2f Bm = {vr[t * 16 + j], vr[D_HD + t * 16 + j]};
            acc[t] = __builtin_amdgcn_wmma_f32_16x16x4_f32(
                false, A, false, Bm, (short)0, acc[t], false, false);
        }
    }

    // D-matrix row M=0 lives in VGPR 0 of lanes 0..15 (N = lane).
    if (lane < 16) {
        #pragma unroll
        for (int t = 0; t < 4; ++t)
            out[(size_t)n * D_HD + t * 16 + lane] = acc[t][0];
    }
}

extern "C" void kernel_launch(void* const* d_in, const int* in_sizes, int n_in,
                              void* d_out, int out_size, void* d_ws, size_t ws_size,
                              hipStream_t stream) {
    const float*         q    = (const float*)d_in[0];
    const float*         k    = (const float*)d_in[1];
    const float*         v    = (const float*)d_in[2];
    const float*         r    = (const float*)d_in[3];
    const unsigned char* mask = (const unsigned char*)d_in[4];

    const int N = in_sizes[0] / (L_NB * D_HD);   // 65536
    const int B = in_sizes[3] / L_NB;            // 32768

    float* out_p  = (float*)d_out;               // [N,64] then [N,20]
    float* attn_p = out_p + (size_t)N * D_HD;

    const int blocks = (N + WAVES_PER_BLOCK - 1) / WAVES_PER_BLOCK;
    hipLaunchKernelGGL(sdpa_prior_kernel, dim3(blocks), dim3(256), 0, stream,
                       q, k, v, r, mask, out_p, attn_p, N, B);
}